// VanilaGCN_61950608277942
// MI455X (gfx1250) — compile-verified
//
#include <hip/hip_runtime.h>
#include <hip/hip_bf16.h>

typedef float v2f __attribute__((ext_vector_type(2)));
typedef float v8f __attribute__((ext_vector_type(8)));

#define F 128          // F_IN == CONV_H == 128
#define NGRAPH 64
#define LINH 64
#define NCLS 10
#define WT_STRIDE 132  // padded LDS column stride (words); 132 mod 64 = 4 -> conflict-free B loads

static __host__ __device__ inline int cdiv(int a, int b) { return (a + b - 1) / b; }

// ---------------- degree / normalization ----------------
__global__ void k_deg_init(float* __restrict__ deg, int n) {
    int i = blockIdx.x * blockDim.x + threadIdx.x;
    if (i < n) deg[i] = 1.0f;                    // self-loop weight 1
}

__global__ void k_deg_accum(const int* __restrict__ col, const float* __restrict__ ew,
                            float* __restrict__ deg, int ne) {
    int e = blockIdx.x * blockDim.x + threadIdx.x;
    if (e < ne) atomicAdd(&deg[col[e]], ew[e]);
}

__global__ void k_dinv(const float* __restrict__ deg, float* __restrict__ dinv, int n) {
    int i = blockIdx.x * blockDim.x + threadIdx.x;
    if (i < n) {
        float d = deg[i];
        dinv[i] = (d > 0.0f) ? rsqrtf(d) : 0.0f;
    }
}

// ---------------- h = x @ conv_w via V_WMMA_F32_16X16X4_F32 ----------------
// W is staged once per block into LDS, transposed (wT[col][k], stride 132 words),
// so each B fragment (k, k+1 for fixed col) is one conflict-free ds_load_b64.
// One wave computes a 16-row strip of h across all 128 output columns.
// A (16x4 f32): 2 VGPRs; lanes 0-15 hold K={k0,k0+1}, lanes 16-31 K={k0+2,k0+3}.
// C/D (16x16 f32): 8 VGPRs; VGPR r = row r (lanes 0-15) / row 8+r (lanes 16-31).
__global__ void __launch_bounds__(256) k_gemm_wmma(const float* __restrict__ x,
                                                   const float* __restrict__ w,
                                                   float* __restrict__ h, int n_nodes) {
    extern __shared__ float wT[];                 // [128 cols][WT_STRIDE] = 66 KB dynamic LDS

    // stage W transposed: coalesced global reads, one-time LDS stores
    for (int idx = threadIdx.x; idx < F * F; idx += 256) {
        const int k = idx >> 7;                   // row of w (k index)
        const int c = idx & 127;                  // col of w
        wT[c * WT_STRIDE + k] = w[idx];
    }
    __syncthreads();

    const int lane  = threadIdx.x & 31;
    const int wave  = threadIdx.x >> 5;
    const int tileM = blockIdx.x * 8 + wave;      // 16-row tile index (wave-uniform)
    if (tileM * 16 >= n_nodes) return;            // whole-wave exit -> EXEC stays full

    const int l15   = lane & 15;
    const int kHalf = (lane >> 4) * 2;            // 0 or 2
    const int mRow  = tileM * 16 + l15;           // A-row for this lane

    v8f c[8];
    const v8f zero = {};
#pragma unroll
    for (int n = 0; n < 8; ++n) c[n] = zero;

    for (int k0 = 0; k0 < F; k0 += 4) {
        v2f a;
        a.x = x[mRow * F + k0 + kHalf];
        a.y = x[mRow * F + k0 + kHalf + 1];
#pragma unroll
        for (int n = 0; n < 8; ++n) {
            const int col = n * 16 + l15;
            const v2f b = *reinterpret_cast<const v2f*>(&wT[col * WT_STRIDE + k0 + kHalf]);
            c[n] = __builtin_amdgcn_wmma_f32_16x16x4_f32(
                false, a, false, b, (short)0, c[n], false, false);
        }
    }

    const int rowBase = tileM * 16 + ((lane >> 4) << 3);   // +8 for upper half-wave
#pragma unroll
    for (int n = 0; n < 8; ++n) {
        const int col = n * 16 + l15;
#pragma unroll
        for (int r = 0; r < 8; ++r)
            h[(rowBase + r) * F + col] = c[n][r];
    }
}

// ---------------- acc init: bias + self-loop term ----------------
__global__ void k_self_init(const float* __restrict__ h, const float* __restrict__ dinv,
                            const float* __restrict__ bias, float* __restrict__ acc, int n) {
    long long idx = (long long)blockIdx.x * blockDim.x + threadIdx.x;
    int i = (int)(idx >> 5);
    int c = ((int)idx & 31) * 4;
    if (i >= n) return;
    float s = dinv[i];
    s = s * s;                                     // norm of self loop = dinv^2 (weight 1)
    const float4 hv = *reinterpret_cast<const float4*>(h + (long long)i * F + c);
    const float4 bv = *reinterpret_cast<const float4*>(bias + c);
    float4 o;
    o.x = bv.x + s * hv.x;  o.y = bv.y + s * hv.y;
    o.z = bv.z + s * hv.z;  o.w = bv.w + s * hv.w;
    *reinterpret_cast<float4*>(acc + (long long)i * F + c) = o;
}

// ---------------- edge scatter: acc[col] += norm * h[row] ----------------
// 32 lanes per edge, float4 per lane -> one contiguous 512B gather per wave,
// wave-uniform row/col/norm scalars. h/acc both live in the 192MB L2.
__global__ void k_scatter(const int* __restrict__ row, const int* __restrict__ col,
                          const float* __restrict__ ew, const float* __restrict__ dinv,
                          const float* __restrict__ h, float* __restrict__ acc, int ne) {
    long long idx = (long long)blockIdx.x * blockDim.x + threadIdx.x;
    int e = (int)(idx >> 5);
    int c = ((int)idx & 31) * 4;
    if (e >= ne) return;
    int r = row[e];
    int t = col[e];
    float norm = dinv[r] * ew[e] * dinv[t];
    const float4 hv = *reinterpret_cast<const float4*>(h + (long long)r * F + c);
    float* dst = acc + (long long)t * F + c;
    atomicAdd(dst + 0, norm * hv.x);
    atomicAdd(dst + 1, norm * hv.y);
    atomicAdd(dst + 2, norm * hv.z);
    atomicAdd(dst + 3, norm * hv.w);
}

// ---------------- pooled init + relu & segment-max pool ----------------
__global__ void k_pool_init(float* __restrict__ pooled, int n) {
    int i = blockIdx.x * blockDim.x + threadIdx.x;
    if (i < n) pooled[i] = 0.0f;                  // relu output >= 0, so 0 is identity
}

#define NODE_TILE 128
__global__ void __launch_bounds__(F) k_pool(const float* __restrict__ acc,
                                            const int* __restrict__ batch,
                                            float* __restrict__ pooled, int n) {
    const int f  = threadIdx.x;                   // one thread per feature
    const int n0 = blockIdx.x * NODE_TILE;
    const int n1 = (n0 + NODE_TILE < n) ? (n0 + NODE_TILE) : n;
    int curg = -1;
    float m = 0.0f;
    for (int i = n0; i < n1; ++i) {
        int g = batch[i];                         // batch is sorted -> few flushes
        if (g != curg) {
            if (curg >= 0)
                atomicMax((unsigned int*)&pooled[curg * F + f], __float_as_uint(m));
            curg = g;
            m = 0.0f;
        }
        float v = acc[(long long)i * F + f];
        v = v > 0.0f ? v : 0.0f;                  // relu
        m = m > v ? m : v;
    }
    if (curg >= 0)
        atomicMax((unsigned int*)&pooled[curg * F + f], __float_as_uint(m));
}

// ---------------- tiny MLP head: relu(pooled@W1+b1)@W2+b2 ----------------
__global__ void __launch_bounds__(LINH) k_mlp(const float* __restrict__ pooled,
                                              const float* __restrict__ w1, const float* __restrict__ b1,
                                              const float* __restrict__ w2, const float* __restrict__ b2,
                                              float* __restrict__ out) {
    __shared__ float pg[F];
    __shared__ float hid[LINH];
    const int g = blockIdx.x;
    const int t = threadIdx.x;
    pg[t]        = pooled[g * F + t];
    pg[t + LINH] = pooled[g * F + LINH + t];
    __syncthreads();
    float s = b1[t];
    for (int k = 0; k < F; ++k) s += pg[k] * w1[k * LINH + t];
    hid[t] = s > 0.0f ? s : 0.0f;
    __syncthreads();
    if (t < NCLS) {
        float o = b2[t];
        for (int k = 0; k < LINH; ++k) o += hid[k] * w2[k * NCLS + t];
        out[g * NCLS + t] = o;
    }
}

extern "C" void kernel_launch(void* const* d_in, const int* in_sizes, int n_in,
                              void* d_out, int out_size, void* d_ws, size_t ws_size,
                              hipStream_t stream) {
    const float* x      = (const float*)d_in[0];
    const int*   eidx   = (const int*)d_in[1];     // [2, E]: row (src), col (dst)
    const float* ew     = (const float*)d_in[2];
    const int*   batch  = (const int*)d_in[3];
    const float* conv_w = (const float*)d_in[4];
    const float* conv_b = (const float*)d_in[5];
    const float* lin1_w = (const float*)d_in[6];
    const float* lin1_b = (const float*)d_in[7];
    const float* lin2_w = (const float*)d_in[8];
    const float* lin2_b = (const float*)d_in[9];
    float* out = (float*)d_out;

    const int N = in_sizes[0] / F;                 // 100000
    const int E = in_sizes[2];                     // 1600000
    const int* row = eidx;
    const int* col = eidx + E;

    // workspace layout (floats): h[N*F] | acc[N*F] | deg[N] | dinv[N] | pooled[64*F]
    float* h      = (float*)d_ws;
    float* acc    = h + (size_t)N * F;
    float* deg    = acc + (size_t)N * F;
    float* dinv   = deg + N;
    float* pooled = dinv + N;

    // 1) symmetric degree normalization
    k_deg_init<<<cdiv(N, 256), 256, 0, stream>>>(deg, N);
    k_deg_accum<<<cdiv(E, 256), 256, 0, stream>>>(col, ew, deg, E);
    k_dinv<<<cdiv(N, 256), 256, 0, stream>>>(deg, dinv, N);

    // 2) h = x @ conv_w  (fp32 WMMA, 8 waves/block, LDS-staged transposed W)
    const int mTiles = cdiv(N, 16);
    const size_t wtBytes = (size_t)F * WT_STRIDE * sizeof(float);   // 66 KB dynamic LDS
    k_gemm_wmma<<<cdiv(mTiles, 8), 256, wtBytes, stream>>>(x, conv_w, h, N);

    // 3) acc = conv_b + dinv^2 * h  (self loops), then edge scatter
    {
        long long th = (long long)N * 32;
        k_self_init<<<(int)((th + 255) / 256), 256, 0, stream>>>(h, dinv, conv_b, acc, N);
    }
    {
        long long th = (long long)E * 32;
        k_scatter<<<(int)((th + 255) / 256), 256, 0, stream>>>(row, col, ew, dinv, h, acc, E);
    }

    // 4) relu + global max pool per graph
    k_pool_init<<<cdiv(NGRAPH * F, 256), 256, 0, stream>>>(pooled, NGRAPH * F);
    k_pool<<<cdiv(N, NODE_TILE), F, 0, stream>>>(acc, batch, pooled, N);

    // 5) MLP head
    k_mlp<<<NGRAPH, LINH, 0, stream>>>(pooled, lin1_w, lin1_b, lin2_w, lin2_b, out);
}